// MaxPool3d_muti_batch_13537736917667
// MI455X (gfx1250) — compile-verified
//
#include <hip/hip_runtime.h>
#include <hip/hip_bf16.h>

// 3D max pool, kernel=stride=(2,2,2), pad 0.
// x: [B=8, D=64, H=256, W=256] f32  ->  out: [8, 32, 128, 128] f32
// Pure streaming op: 128 MiB read + 16 MiB write, ~6.5 us at 23.3 TB/s.
// Strategy: one thread = 2 adjacent output W-elements.
//   Loads 4x float4 (global_load_b128, lane-contiguous -> 512B/wave/load),
//   14 max ops, one float2 store (global_store_b64). Non-temporal hints:
//   data is single-use, no reason to occupy cache.

typedef __attribute__((ext_vector_type(4))) float v4f;
typedef __attribute__((ext_vector_type(2))) float v2f;

#define B_N    8
#define D_IN   64
#define H_IN   256
#define W_IN   256
#define D_OUT  32
#define H_OUT  128
#define W_OUT  128
#define W_PAIR (W_OUT / 2)   // 64 thread-tiles per output row

__global__ __launch_bounds__(256) void
maxpool3d_222_kernel(const float* __restrict__ x,
                     float* __restrict__ out,
                     int total_pairs) {
    int tid = blockIdx.x * blockDim.x + threadIdx.x;
    if (tid >= total_pairs) return;

    // Decompose (all power-of-two -> shifts/masks)
    int wp = tid & (W_PAIR - 1);        // which pair of output columns
    int t1 = tid >> 6;
    int ho = t1 & (H_OUT - 1);          // output row
    int t2 = t1 >> 7;
    int dd = t2 & (D_OUT - 1);          // output depth slice
    int b  = t2 >> 5;                   // batch

    const size_t row   = (size_t)W_IN;          // 256 floats
    const size_t plane = (size_t)H_IN * W_IN;   // 65536 floats

    // Input window origin: d = 2*dd, h = 2*ho, w = 4*wp (4 input cols -> 2 outputs)
    size_t base = ((size_t)b * D_IN + (size_t)(dd << 1)) * plane
                + (size_t)(ho << 1) * row
                + (size_t)(wp << 2);

    // 4 contributing input rows: (d,h), (d,h+1), (d+1,h), (d+1,h+1)
    v4f r00 = __builtin_nontemporal_load((const v4f*)(x + base));
    v4f r01 = __builtin_nontemporal_load((const v4f*)(x + base + row));
    v4f r10 = __builtin_nontemporal_load((const v4f*)(x + base + plane));
    v4f r11 = __builtin_nontemporal_load((const v4f*)(x + base + plane + row));

    // Vertical (H,D) reduction, then horizontal (W) pairwise reduction.
    v4f m;
    m.x = fmaxf(fmaxf(r00.x, r01.x), fmaxf(r10.x, r11.x));
    m.y = fmaxf(fmaxf(r00.y, r01.y), fmaxf(r10.y, r11.y));
    m.z = fmaxf(fmaxf(r00.z, r01.z), fmaxf(r10.z, r11.z));
    m.w = fmaxf(fmaxf(r00.w, r01.w), fmaxf(r10.w, r11.w));

    v2f res;
    res.x = fmaxf(m.x, m.y);
    res.y = fmaxf(m.z, m.w);

    size_t obase = (((size_t)b * D_OUT + (size_t)dd) * H_OUT + (size_t)ho) * W_OUT
                 + (size_t)(wp << 1);
    __builtin_nontemporal_store(res, (v2f*)(out + obase));
}

extern "C" void kernel_launch(void* const* d_in, const int* in_sizes, int n_in,
                              void* d_out, int out_size, void* d_ws, size_t ws_size,
                              hipStream_t stream) {
    (void)in_sizes; (void)n_in; (void)d_ws; (void)ws_size; (void)out_size;

    const float* x = (const float*)d_in[0];
    float* out     = (float*)d_out;

    const int total_pairs = B_N * D_OUT * H_OUT * W_PAIR;  // 2,097,152 threads
    const int block = 256;                                  // 8 wave32 waves
    const int grid  = (total_pairs + block - 1) / block;    // 8192 blocks

    maxpool3d_222_kernel<<<grid, block, 0, stream>>>(x, out, total_pairs);
}